// Decoder_128849019075
// MI455X (gfx1250) — compile-verified
//
#include <hip/hip_runtime.h>
#include <hip/hip_bf16.h>
#include <math.h>

typedef float v2f __attribute__((ext_vector_type(2)));
typedef float v8f __attribute__((ext_vector_type(8)));

#define DEVINL __device__ __forceinline__

// ---------------------------------------------------------------------------
// WMMA f32 16x16x4 helper (fp32 matrix core path; exact vs f32 reference)
// ---------------------------------------------------------------------------
DEVINL v8f wmma4(v2f a, v2f b, v8f c) {
  return __builtin_amdgcn_wmma_f32_16x16x4_f32(
      /*neg_a=*/false, a, /*neg_b=*/false, b,
      /*c_mod=*/(short)0, c, /*reuse_a=*/false, /*reuse_b=*/false);
}

// C += A[M,K] * W^T, where W is row-major [N,K] (ldw = row stride).
// Fragment layout (V_WMMA_F32_16X16X4_F32):
//   A: lane<16 holds (m, k+0..1), lane>=16 holds (m, k+2..3)
//   B mirrors with n instead of m.
// Each lane loads exactly its 8 bytes (b64) at row + k + 2*hb: no selects,
// both halves jointly cover k..k+3 contiguously -> full cacheline use.
// Requires K % 32 == 0 (true for all call sites: 1536, 1024).
DEVINL void gemm_nt(v8f& acc, const float* __restrict__ A, int lda,
                    const float* __restrict__ W, int ldw, int K,
                    int m0, int n0, int idx, int hb) {
  const float* ap = A + (size_t)(m0 + idx) * lda + 2 * hb;
  const float* wp = W + (size_t)(n0 + idx) * ldw + 2 * hb;
  for (int k = 0; k < K; k += 32) {
    // keep the per-row HBM weight stream primed (~2KB ahead)
    __builtin_prefetch(wp + k + 512, 0, 0);
#pragma unroll
    for (int kk = 0; kk < 32; kk += 4) {
      v2f a = *(const v2f*)(ap + k + kk);
      v2f b = *(const v2f*)(wp + k + kk);
      acc = wmma4(a, b, acc);
    }
  }
}

// C += A[M,K] * B, where B is row-major [K,N] (ldb = row stride).
DEVINL void gemm_nn(v8f& acc, const float* __restrict__ A, int lda,
                    const float* __restrict__ B, int ldb, int K,
                    int m0, int n0, int idx, int hb) {
  const float* ap = A + (size_t)(m0 + idx) * lda + 2 * hb;
  const float* bp = B + (size_t)2 * hb * ldb + (n0 + idx);
  for (int k = 0; k < K; k += 32) {
#pragma unroll
    for (int kk = 0; kk < 32; kk += 4) {
      v2f a = *(const v2f*)(ap + k + kk);
      v2f b;
      b.x = bp[(size_t)(k + kk) * ldb];
      b.y = bp[(size_t)(k + kk + 1) * ldb];
      acc = wmma4(a, b, acc);
    }
  }
}

// acc[v] corresponds to C[m0 + v + 8*half][n0 + idx]
DEVINL void store_tile(float* __restrict__ C, int ldc, v8f acc,
                       int m0, int n0, int idx, int hb) {
  int n = n0 + idx;
#pragma unroll
  for (int v = 0; v < 8; ++v)
    C[(size_t)(m0 + v + 8 * hb) * ldc + n] = acc[v];
}

// ---------------------------------------------------------------------------
// Kernel 0: x = [emb[inputs], context]  -> [64, 1536]
// ---------------------------------------------------------------------------
__global__ void xcat_kernel(const int* __restrict__ inputs,
                            const float* __restrict__ emb,
                            const float* __restrict__ context,
                            float* __restrict__ xcat) {
  int t = blockIdx.x * blockDim.x + threadIdx.x;
  if (t >= 64 * 1536) return;
  int b = t / 1536;
  int k = t - b * 1536;
  float v = (k < 512) ? emb[(size_t)inputs[b] * 512 + k]
                      : context[(size_t)b * 1024 + (k - 512)];
  xcat[t] = v;
}

// ---------------------------------------------------------------------------
// LSTM gates: out[64,4096] = A1@W1^T + bias1 + A2@W2^T + bias2
// grid: 128 blocks x 256 thr (8 waves) ; wave -> (m_tile = wid&3, n_tile)
// ---------------------------------------------------------------------------
__global__ void gates_kernel(const float* __restrict__ A1, int lda1, int K1,
                             const float* __restrict__ W1,
                             const float* __restrict__ A2, int lda2, int K2,
                             const float* __restrict__ W2,
                             const float* __restrict__ bias1,
                             const float* __restrict__ bias2,
                             float* __restrict__ out) {
  int lane = threadIdx.x & 31, idx = lane & 15, hb = lane >> 4;
  int wid = threadIdx.x >> 5;
  int m0 = (wid & 3) * 16;
  int n0 = (blockIdx.x * 2 + (wid >> 2)) * 16;
  float bs = bias1[n0 + idx] + bias2[n0 + idx];
  v8f acc;
#pragma unroll
  for (int v = 0; v < 8; ++v) acc[v] = bs;
  gemm_nt(acc, A1, lda1, W1, K1, K1, m0, n0, idx, hb);
  gemm_nt(acc, A2, lda2, W2, K2, K2, m0, n0, idx, hb);
  store_tile(out, 4096, acc, m0, n0, idx, hb);
}

// ---------------------------------------------------------------------------
// LSTM cell elementwise (gate order i,f,g,o per PyTorch)
// ---------------------------------------------------------------------------
DEVINL float sigf(float x) { return 1.0f / (1.0f + expf(-x)); }

__global__ void cell_kernel(const float* __restrict__ gates,
                            const float* __restrict__ c_in,
                            float* __restrict__ h_out,
                            float* __restrict__ c_out,
                            float* __restrict__ h_ws) {
  int t = blockIdx.x * blockDim.x + threadIdx.x;  // 65536
  int b = t >> 10, d = t & 1023;
  const float* g = gates + (size_t)b * 4096;
  float ig = sigf(g[d]);
  float fg = sigf(g[d + 1024]);
  float gg = tanhf(g[d + 2048]);
  float og = sigf(g[d + 3072]);
  float c = fg * c_in[t] + ig * gg;
  float h = og * tanhf(c);
  c_out[t] = c;
  h_out[t] = h;
  h_ws[t] = h;
}

// ---------------------------------------------------------------------------
// u[64,1024] = ro @ attn_W   (attn_W row-major [K=1024, N=1024], NN)
// grid 32 x 256
// ---------------------------------------------------------------------------
__global__ void u_kernel(const float* __restrict__ ro,
                         const float* __restrict__ attn_W,
                         float* __restrict__ u) {
  int lane = threadIdx.x & 31, idx = lane & 15, hb = lane >> 4;
  int wid = threadIdx.x >> 5;
  int m0 = (wid & 3) * 16;
  int n0 = (blockIdx.x * 2 + (wid >> 2)) * 16;
  v8f acc;
#pragma unroll
  for (int v = 0; v < 8; ++v) acc[v] = 0.0f;
  gemm_nn(acc, ro, 1024, attn_W, 1024, 1024, m0, n0, idx, hb);
  store_tile(u, 1024, acc, m0, n0, idx, hb);
}

// beta[b] = dot(ro[b], attn_b)   grid 64 x 32
__global__ void beta_kernel(const float* __restrict__ ro,
                            const float* __restrict__ attn_b,
                            float* __restrict__ beta) {
  int b = blockIdx.x, lane = threadIdx.x;
  float s = 0.0f;
  for (int k = lane; k < 1024; k += 32) s += ro[(size_t)b * 1024 + k] * attn_b[k];
#pragma unroll
  for (int off = 16; off; off >>= 1) s += __shfl_xor(s, off, 32);
  if (lane == 0) beta[b] = s;
}

// scores[s*64+b] = E[s,b,:]·u[b,:] + beta[b]   grid 1024 x 256 (wave per pair)
__global__ void scores_kernel(const float* __restrict__ E,
                              const float* __restrict__ u,
                              const float* __restrict__ beta,
                              float* __restrict__ scores) {
  int pair = blockIdx.x * 8 + (threadIdx.x >> 5);  // pair = s*64 + b
  int lane = threadIdx.x & 31;
  int b = pair & 63;
  const float* e = E + (size_t)pair * 1024;
  const float* ub = u + (size_t)b * 1024;
  float acc = 0.0f;
  for (int k = lane * 4; k < 1024; k += 128) {
    float4 e4 = *(const float4*)(e + k);
    float4 u4 = *(const float4*)(ub + k);
    acc += e4.x * u4.x + e4.y * u4.y + e4.z * u4.z + e4.w * u4.w;
  }
#pragma unroll
  for (int off = 16; off; off >>= 1) acc += __shfl_xor(acc, off, 32);
  if (lane == 0) scores[pair] = acc + beta[b];
}

// log_softmax over the BATCH dim (reference quirk).  grid 128 x 64
__global__ void lsm_batch_kernel(const float* __restrict__ scores,
                                 float* __restrict__ logw,
                                 float* __restrict__ attnw_out) {
  __shared__ float red[64];
  int s = blockIdx.x, b = threadIdx.x;
  float v = scores[s * 64 + b];
  red[b] = v; __syncthreads();
  for (int off = 32; off; off >>= 1) {
    if (b < off) red[b] = fmaxf(red[b], red[b + off]);
    __syncthreads();
  }
  float m = red[0]; __syncthreads();
  red[b] = expf(v - m); __syncthreads();
  for (int off = 32; off; off >>= 1) {
    if (b < off) red[b] += red[b + off];
    __syncthreads();
  }
  float lw = v - (m + logf(red[0]));
  logw[s * 64 + b] = lw;
  attnw_out[b * 128 + s] = lw;  // attn_weights[b,0,s]
}

// new_context[b,d] = sum_s logw[s,b] * E[s,b,d]   grid 256 x 256
__global__ void ctx_kernel(const float* __restrict__ logw,
                           const float* __restrict__ E,
                           float* __restrict__ ctx_out) {
  __shared__ float lw[128];
  int t = blockIdx.x * 256 + threadIdx.x;  // 65536
  int b = t >> 10, d = t & 1023;           // whole block shares b (256 | 1024)
  if (threadIdx.x < 128) lw[threadIdx.x] = logw[threadIdx.x * 64 + b];
  __syncthreads();
  float acc = 0.0f;
#pragma unroll 4
  for (int s = 0; s < 128; ++s)
    acc += lw[s] * E[((size_t)s * 64 + b) * 1024 + d];
  ctx_out[t] = acc;
}

// logits[64,32000] = [ro|ctx] @ out_W^T + out_b   grid 1000 x 256
__global__ void logits_kernel(const float* __restrict__ ro,
                              const float* __restrict__ ctx,
                              const float* __restrict__ out_W,
                              const float* __restrict__ out_b,
                              float* __restrict__ logits) {
  int lane = threadIdx.x & 31, idx = lane & 15, hb = lane >> 4;
  int wid = threadIdx.x >> 5;
  int m0 = (wid & 3) * 16;
  int n0 = (blockIdx.x * 2 + (wid >> 2)) * 16;
  float bs = out_b[n0 + idx];
  v8f acc;
#pragma unroll
  for (int v = 0; v < 8; ++v) acc[v] = bs;
  gemm_nt(acc, ro, 1024, out_W, 2048, 1024, m0, n0, idx, hb);
  gemm_nt(acc, ctx, 1024, out_W + 1024, 2048, 1024, m0, n0, idx, hb);
  store_tile(logits, 32000, acc, m0, n0, idx, hb);
}

// log_softmax over V per batch row.  grid 64 x 256
__global__ void lsm_v_kernel(const float* __restrict__ logits,
                             float* __restrict__ out) {
  __shared__ float red[256];
  int b = blockIdx.x, t = threadIdx.x;
  const float* row = logits + (size_t)b * 32000;
  float m = -INFINITY;
  for (int v = t; v < 32000; v += 256) m = fmaxf(m, row[v]);
  red[t] = m; __syncthreads();
  for (int off = 128; off; off >>= 1) {
    if (t < off) red[t] = fmaxf(red[t], red[t + off]);
    __syncthreads();
  }
  m = red[0]; __syncthreads();
  float s = 0.0f;
  for (int v = t; v < 32000; v += 256) s += expf(row[v] - m);
  red[t] = s; __syncthreads();
  for (int off = 128; off; off >>= 1) {
    if (t < off) red[t] += red[t + off];
    __syncthreads();
  }
  float lse = m + logf(red[0]);
  for (int v = t; v < 32000; v += 256)
    out[(size_t)b * 32000 + v] = row[v] - lse;
}

// ---------------------------------------------------------------------------
extern "C" void kernel_launch(void* const* d_in, const int* in_sizes, int n_in,
                              void* d_out, int out_size, void* d_ws, size_t ws_size,
                              hipStream_t stream) {
  (void)in_sizes; (void)n_in; (void)out_size; (void)ws_size;
  const float* h0      = (const float*)d_in[0];   // [2,64,1024]
  const float* c0      = (const float*)d_in[1];   // [2,64,1024]
  const float* context = (const float*)d_in[2];   // [64,1024]
  const float* E       = (const float*)d_in[3];   // [128,64,1024]
  const int*   inputs  = (const int*)d_in[4];     // [64]
  const float* emb     = (const float*)d_in[5];   // [32000,512]
  const float* attn_W  = (const float*)d_in[6];   // [1024,1024]
  const float* attn_b  = (const float*)d_in[7];   // [1024]
  const float* W_ih0   = (const float*)d_in[8];   // [4096,1536]
  const float* W_hh0   = (const float*)d_in[9];   // [4096,1024]
  const float* b_ih0   = (const float*)d_in[10];
  const float* b_hh0   = (const float*)d_in[11];
  const float* W_ih1   = (const float*)d_in[12];  // [4096,1024]
  const float* W_hh1   = (const float*)d_in[13];  // [4096,1024]
  const float* b_ih1   = (const float*)d_in[14];
  const float* b_hh1   = (const float*)d_in[15];
  const float* out_W   = (const float*)d_in[16];  // [32000,2048]
  const float* out_b   = (const float*)d_in[17];

  float* out = (float*)d_out;
  // output tuple layout (flat, in return order)
  const size_t OUT_LOGP  = 0;                 // [64,32000]
  const size_t OUT_CTX   = 2048000;           // [64,1024]
  const size_t OUT_H     = OUT_CTX + 65536;   // [2,64,1024]
  const size_t OUT_C     = OUT_H + 131072;    // [2,64,1024]
  const size_t OUT_ATTNW = OUT_C + 131072;    // [64,1,128]

  // workspace layout (floats)
  float* ws     = (float*)d_ws;
  float* xcat   = ws;                  // 64*1536
  float* gates  = xcat + 98304;        // 64*4096
  float* h0n    = gates + 262144;      // 64*1024
  float* ro     = h0n + 65536;         // 64*1024
  float* u      = ro + 65536;          // 64*1024
  float* beta   = u + 65536;           // 64
  float* scores = beta + 64;           // 128*64
  float* logw   = scores + 8192;       // 128*64
  float* logits = logw + 8192;         // 64*32000

  // 1) x = [emb[inputs], context]
  xcat_kernel<<<(98304 + 255) / 256, 256, 0, stream>>>(inputs, emb, context, xcat);
  // 2) layer-0 gates + cell
  gates_kernel<<<128, 256, 0, stream>>>(xcat, 1536, 1536, W_ih0,
                                        h0, 1024, 1024, W_hh0,
                                        b_ih0, b_hh0, gates);
  cell_kernel<<<256, 256, 0, stream>>>(gates, c0, out + OUT_H, out + OUT_C, h0n);
  // 3) layer-1 gates + cell (ro = h1n)
  gates_kernel<<<128, 256, 0, stream>>>(h0n, 1024, 1024, W_ih1,
                                        h0 + 65536, 1024, 1024, W_hh1,
                                        b_ih1, b_hh1, gates);
  cell_kernel<<<256, 256, 0, stream>>>(gates, c0 + 65536,
                                       out + OUT_H + 65536, out + OUT_C + 65536, ro);
  // 4) attention (factored): u = ro@attn_W ; beta = ro·attn_b ; scores = E·u + beta
  u_kernel<<<32, 256, 0, stream>>>(ro, attn_W, u);
  beta_kernel<<<64, 32, 0, stream>>>(ro, attn_b, beta);
  scores_kernel<<<1024, 256, 0, stream>>>(E, u, beta, scores);
  lsm_batch_kernel<<<128, 64, 0, stream>>>(scores, logw, out + OUT_ATTNW);
  ctx_kernel<<<256, 256, 0, stream>>>(logw, E, out + OUT_CTX);
  // 5) output projection + log-softmax
  logits_kernel<<<1000, 256, 0, stream>>>(ro, out + OUT_CTX, out_W, out_b, logits);
  lsm_v_kernel<<<64, 256, 0, stream>>>(logits, out + OUT_LOGP);
}